// GaussianMatchingLoss_77249281786043
// MI455X (gfx1250) — compile-verified
//
#include <hip/hip_runtime.h>

typedef __attribute__((ext_vector_type(2))) float v2f;
typedef __attribute__((ext_vector_type(8))) float v8f;

#define B_BATCH 8
#define NP 4096
#define NT 8192
#define NC 14
#define EPSQ 1e-8f

// ---- ds_swizzle helpers (group-of-32 xor mode: offset = xor<<10 | and=0x1f) ----
template <int IMM>
__device__ __forceinline__ float swzf(float v) {
  return __int_as_float(__builtin_amdgcn_ds_swizzle(__float_as_int(v), IMM));
}
template <int IMM>
__device__ __forceinline__ int swzi(int v) {
  return __builtin_amdgcn_ds_swizzle(v, IMM);
}
template <int MASK>
__device__ __forceinline__ void argmin_step(float& v, int& idx) {
  constexpr int IMM = (MASK << 10) | 0x1f;
  float ov = swzf<IMM>(v);
  int   oi = swzi<IMM>(idx);
  if (ov < v || (ov == v && oi < idx)) { v = ov; idx = oi; }
}

// For every row of A (xyz in cols 0..2), find argmin over rows of B of
//   |b|^2 - 2*a.b   (== squared distance minus the row-constant |a|^2)
// One wave owns TWO 16-row A-tiles (32 rows): the B tile load and |b|^2 are
// amortized over both, and the two independent WMMAs pipeline against the
// argmin VALU chain (XDL/VALU co-execution).
__global__ __launch_bounds__(256)
void argmin_rows_kernel(const float* __restrict__ Arows,
                        const float* __restrict__ Bcols,
                        int* __restrict__ out_idx,
                        int NA, int NB) {
  const int b    = blockIdx.y;
  const int tid  = threadIdx.x;
  const int wave = tid >> 5;
  const int lane = tid & 31;
  const int half = lane >> 4;   // 0: K0/K1 side, 1: K2/pad side (A layout 16x4 f32)
  const int l    = lane & 15;
  const int i0   = blockIdx.x * 256 + wave * 32;   // two 16-row tiles per wave

  const float* __restrict__ A  = Arows + (size_t)b * NA * NC;
  const float* __restrict__ Bp = Bcols + (size_t)b * NB * NC;

  // A tiles: 16 rows x K=4 (xyz,0) each
  v2f a0, a1;
  {
    const int r0 = i0 + l;
    const float x0 = A[r0 * NC + 0], y0 = A[r0 * NC + 1], z0 = A[r0 * NC + 2];
    a0.x = half ? z0 : x0;
    a0.y = half ? 0.0f : y0;
    const int r1 = i0 + 16 + l;
    const float x1 = A[r1 * NC + 0], y1 = A[r1 * NC + 1], z1 = A[r1 * NC + 2];
    a1.x = half ? z1 : x1;
    a1.y = half ? 0.0f : y1;
  }

  float best0[8], best1[8];
  int   bidx0[8], bidx1[8];
#pragma unroll
  for (int r = 0; r < 8; ++r) {
    best0[r] = 3.0e38f; bidx0[r] = 0;
    best1[r] = 3.0e38f; bidx1[r] = 0;
  }

#pragma unroll 2
  for (int j0 = 0; j0 < NB; j0 += 16) {
    const int col = j0 + l;                    // lanes l and l+16 share this column
    const float tx = Bp[col * NC + 0];
    const float ty = Bp[col * NC + 1];
    const float tz = Bp[col * NC + 2];
    const float tn = tx * tx + ty * ty + tz * tz;
    v2f bm;
    bm.x = half ? tz : tx;
    bm.y = half ? 0.0f : ty;

    v8f c0 = {0.f, 0.f, 0.f, 0.f, 0.f, 0.f, 0.f, 0.f};
    v8f c1 = {0.f, 0.f, 0.f, 0.f, 0.f, 0.f, 0.f, 0.f};
    // c[r] = dot(A_row{M}, B_col{N}), M = r + 8*half (+16 for tile 1), N = l
    c0 = __builtin_amdgcn_wmma_f32_16x16x4_f32(false, a0, false, bm,
                                               (short)0, c0, false, false);
    c1 = __builtin_amdgcn_wmma_f32_16x16x4_f32(false, a1, false, bm,
                                               (short)0, c1, false, false);
#pragma unroll
    for (int r = 0; r < 8; ++r) {
      const float m0 = fmaf(-2.0f, c0[r], tn);
      if (m0 < best0[r]) { best0[r] = m0; bidx0[r] = col; }  // strict < keeps first
      const float m1 = fmaf(-2.0f, c1[r], tn);
      if (m1 < best1[r]) { best1[r] = m1; bidx1[r] = col; }
    }
  }

  // reduce across the 16 lanes of each half (xor masks stay inside the half)
#pragma unroll
  for (int r = 0; r < 8; ++r) {
    argmin_step<1>(best0[r], bidx0[r]);
    argmin_step<2>(best0[r], bidx0[r]);
    argmin_step<4>(best0[r], bidx0[r]);
    argmin_step<8>(best0[r], bidx0[r]);
    argmin_step<1>(best1[r], bidx1[r]);
    argmin_step<2>(best1[r], bidx1[r]);
    argmin_step<4>(best1[r], bidx1[r]);
    argmin_step<8>(best1[r], bidx1[r]);
  }

  if (l == 0) {
    int* o = out_idx + (size_t)b * NA + i0 + half * 8;
#pragma unroll
    for (int r = 0; r < 8; ++r) o[r] = bidx0[r];
#pragma unroll
    for (int r = 0; r < 8; ++r) o[16 + r] = bidx1[r];
  }
}

// Per-batch gathered losses -> partials[b*7 + {batch,pos,scale,rot,color,opac,cov}]
__global__ __launch_bounds__(256)
void loss_kernel(const float* __restrict__ pred,
                 const float* __restrict__ targ,
                 const int* __restrict__ fwd_idx,
                 const int* __restrict__ bwd_idx,
                 float* __restrict__ partials) {
  const int b   = blockIdx.x;
  const int tid = threadIdx.x;
  const float* pb = pred + (size_t)b * NP * NC;
  const float* tb = targ + (size_t)b * NT * NC;

  float acc[9];
#pragma unroll
  for (int k = 0; k < 9; ++k) acc[k] = 0.0f;

  for (int i = tid; i < NP; i += 256) {
    const float* p  = pb + (size_t)i * NC;
    const float* tm = tb + (size_t)fwd_idx[(size_t)b * NP + i] * NC;
    float d;
#pragma unroll
    for (int c = 0; c < 3; ++c) { d = p[c] - tm[c]; acc[0] += d * d; }          // pos
#pragma unroll
    for (int c = 3; c < 6; ++c) { d = p[c] - tm[c]; acc[1] += d * d; }          // scale
#pragma unroll
    for (int c = 10; c < 13; ++c) { d = p[c] - tm[c]; acc[3] += d * d; }        // color
    d = p[13] - tm[13]; acc[4] += d * d;                                        // opacity
    // rotation: |dot(q_p_hat, q_t_hat)|
    float np = sqrtf(p[6]*p[6] + p[7]*p[7] + p[8]*p[8] + p[9]*p[9]);
    float nt = sqrtf(tm[6]*tm[6] + tm[7]*tm[7] + tm[8]*tm[8] + tm[9]*tm[9]);
    np = fmaxf(np, EPSQ);
    nt = fmaxf(nt, EPSQ);
    const float dq = p[6]*tm[6] + p[7]*tm[7] + p[8]*tm[8] + p[9]*tm[9];
    acc[2] += fabsf(dq / (np * nt));
  }

  for (int j = tid; j < NT; j += 256) {
    const float* t  = tb + (size_t)j * NC;
    const float* pm = pb + (size_t)bwd_idx[(size_t)b * NT + j] * NC;
    float d;
#pragma unroll
    for (int c = 0; c < 3; ++c) { d = t[c] - pm[c]; acc[5] += d * d; }
#pragma unroll
    for (int c = 3; c < 6; ++c) { d = t[c] - pm[c]; acc[6] += d * d; }
#pragma unroll
    for (int c = 10; c < 13; ++c) { d = t[c] - pm[c]; acc[7] += d * d; }
    d = t[13] - pm[13]; acc[8] += d * d;
  }

  __shared__ float sm[256 * 9];
#pragma unroll
  for (int k = 0; k < 9; ++k) sm[tid * 9 + k] = acc[k];
  __syncthreads();
  for (int s = 128; s > 0; s >>= 1) {
    if (tid < s) {
#pragma unroll
      for (int k = 0; k < 9; ++k) sm[tid * 9 + k] += sm[(tid + s) * 9 + k];
    }
    __syncthreads();
  }

  if (tid == 0) {
    const float inv_p3 = 1.0f / (NP * 3.0f), inv_p = 1.0f / NP;
    const float inv_t3 = 1.0f / (NT * 3.0f), inv_t = 1.0f / NT;
    const float pos = sm[0] * inv_p3;
    const float scl = sm[1] * inv_p3;
    const float rot = 1.0f - sm[2] * inv_p;
    const float col = sm[3] * inv_p3;
    const float opa = sm[4] * inv_p;
    const float cov = 2.0f * sm[5] * inv_t3 + 0.5f * sm[6] * inv_t3 +
                      0.5f * sm[7] * inv_t3 + 2.0f * sm[8] * inv_t;
    const float bl = 10.0f * pos + 5.0f * scl + 2.0f * rot +
                     5.0f * col + 3.0f * opa + 1.0f * cov;
    float* o = partials + b * 7;
    o[0] = bl; o[1] = pos; o[2] = scl; o[3] = rot; o[4] = col; o[5] = opa; o[6] = cov;
  }
}

__global__ void finalize_kernel(const float* __restrict__ partials,
                                float* __restrict__ out) {
  if (threadIdx.x == 0 && blockIdx.x == 0) {
    float tot = 0.0f;
    for (int b = 0; b < B_BATCH; ++b) tot += partials[b * 7];
    out[0] = tot * (1.0f / B_BATCH);
    for (int k = 1; k < 7; ++k) out[k] = partials[(B_BATCH - 1) * 7 + k];
  }
}

extern "C" void kernel_launch(void* const* d_in, const int* in_sizes, int n_in,
                              void* d_out, int out_size, void* d_ws, size_t ws_size,
                              hipStream_t stream) {
  (void)in_sizes; (void)n_in; (void)out_size; (void)ws_size;
  const float* pred = (const float*)d_in[0];   // (8, 4096, 14) f32
  const float* targ = (const float*)d_in[1];   // (8, 8192, 14) f32

  // workspace layout: fwd_idx[8*4096] | bwd_idx[8*8192] | partials[8*7]
  int*   fwd_idx  = (int*)d_ws;
  int*   bwd_idx  = fwd_idx + B_BATCH * NP;
  float* partials = (float*)(bwd_idx + B_BATCH * NT);

  const dim3 blk(256);
  // fwd: for each pred row, argmin over target rows (32 rows per wave)
  argmin_rows_kernel<<<dim3(NP / 256, B_BATCH), blk, 0, stream>>>(
      pred, targ, fwd_idx, NP, NT);
  // bwd: for each target row, argmin over pred rows
  argmin_rows_kernel<<<dim3(NT / 256, B_BATCH), blk, 0, stream>>>(
      targ, pred, bwd_idx, NT, NP);
  loss_kernel<<<dim3(B_BATCH), blk, 0, stream>>>(pred, targ, fwd_idx, bwd_idx, partials);
  finalize_kernel<<<1, 32, 0, stream>>>(partials, (float*)d_out);
}